// GCN_90692529423019
// MI455X (gfx1250) — compile-verified
//
#include <hip/hip_runtime.h>
#include <math.h>

// ---------------------------------------------------------------------------
// ChebConv GNN for MI455X (gfx1250, wave32).
//  - propagate: edge-parallel gather * w + f32 atomic scatter (L2-resident)
//  - dense einsums: V_WMMA_F32_16X16X4_F32, one 16x16 C tile per wave,
//    weights preloaded in registers, Cheb recurrence fused into A-frag load.
// ---------------------------------------------------------------------------

typedef __attribute__((ext_vector_type(2))) float v2f;
typedef __attribute__((ext_vector_type(8))) float v8f;

#define BATCH 4
#define FEAT  64   // F_IN == H == 64

// ---------------- utility fills ----------------
__global__ void fill0_vec4(float4* __restrict__ p, int n4) {
    int i = blockIdx.x * blockDim.x + threadIdx.x;
    if (i < n4) p[i] = make_float4(0.f, 0.f, 0.f, 0.f);
}
__global__ void fill0_scalar(float* __restrict__ p, int n) {
    int i = blockIdx.x * blockDim.x + threadIdx.x;
    if (i < n) p[i] = 0.f;
}

// ---------------- edge weights ----------------
__global__ void deg_kernel(const int* __restrict__ src, const int* __restrict__ dst,
                           float* __restrict__ deg, int E) {
    int e = blockIdx.x * blockDim.x + threadIdx.x;
    if (e < E) {
        int s = src[e];
        if (s != dst[e]) atomicAdd(&deg[s], 1.0f);
    }
}
__global__ void dis_kernel(float* __restrict__ deg, int N) {
    int i = blockIdx.x * blockDim.x + threadIdx.x;
    if (i < N) { float d = deg[i]; deg[i] = (d > 0.f) ? rsqrtf(d) : 0.f; }
}
__global__ void edgew_kernel(const int* __restrict__ src, const int* __restrict__ dst,
                             const float* __restrict__ dis, float* __restrict__ w, int E) {
    int e = blockIdx.x * blockDim.x + threadIdx.x;
    if (e < E) {
        int s = src[e], d = dst[e];
        w[e] = (s != d) ? (-dis[s] * dis[d]) : 0.f;
    }
}

// ---------------- transpose x[B,F,N] -> h[B*N, F] (LDS tiled) ----------------
__global__ __launch_bounds__(256)
void transpose_kernel(const float* __restrict__ x, float* __restrict__ h, int N) {
    __shared__ float tile[FEAT][33];
    const int n0 = blockIdx.x * 32;
    const int b  = blockIdx.y;
    const int t  = threadIdx.x;
    #pragma unroll
    for (int it = 0; it < 8; ++it) {           // coalesced along n
        int f = it * 8 + (t >> 5);
        int j = t & 31;
        int n = n0 + j;
        tile[f][j] = (n < N) ? x[((size_t)b * FEAT + f) * N + n] : 0.f;
    }
    __syncthreads();
    #pragma unroll
    for (int it = 0; it < 8; ++it) {           // coalesced along f
        int j = it * 4 + (t >> 6);
        int f = t & 63;
        int n = n0 + j;
        if (n < N) h[((size_t)b * N + n) * FEAT + f] = tile[f][j];
    }
}

// ---------------- propagate: P[b,dst,:] += w[e] * X[b,src,:] ----------------
__global__ __launch_bounds__(256)
void propagate_kernel(const float* __restrict__ X, const int* __restrict__ src,
                      const int* __restrict__ dst, const float* __restrict__ w,
                      float* __restrict__ P, int N, int E) {
    long long total = (long long)E * BATCH * 16;            // 16 lanes * float4 = 64 feats
    long long t = (long long)blockIdx.x * blockDim.x + threadIdx.x;
    if (t >= total) return;
    int f4   = (int)(t & 15) * 4;
    long long pr = t >> 4;
    int b = (int)(pr & 3);
    int e = (int)(pr >> 2);
    float we = w[e];
    if (we == 0.f) return;                                   // self-loops contribute 0
    int s = src[e], d = dst[e];
    const float4 v = *(const float4*)(X + ((size_t)b * N + s) * FEAT + f4);
    float* pp = P + ((size_t)b * N + d) * FEAT + f4;
    atomicAdd(pp + 0, v.x * we);
    atomicAdd(pp + 1, v.y * we);
    atomicAdd(pp + 2, v.z * we);
    atomicAdd(pp + 3, v.w * we);
}

// ---------------- fused Cheb layer (WMMA f32 16x16x4) ----------------
// Hout = ELU( Tx0*W[0] + Tx1*W[1] + (2*P2 - Tx0)*W[2] + bias )
// One wave per 16x16 output tile. blockIdx.x in [0,4): column tile n0.
__global__ __launch_bounds__(128)
void cheb_layer_kernel(const float* __restrict__ Tx0, const float* __restrict__ Tx1,
                       const float* __restrict__ P2,  const float* __restrict__ W,  // [3][64][64]
                       const float* __restrict__ bias, float* __restrict__ Hout,
                       int numRowTiles) {
    const int lane = threadIdx.x & 31;
    const int wave = threadIdx.x >> 5;
    const int half = lane >> 4;       // 0: K pair {0,1}, 1: K pair {2,3}
    const int l16  = lane & 15;
    const int n0   = blockIdx.x * 16;

    // Preload weight B-fragments for all 16 K-steps of the three Cheb matrices.
    // B layout (4x16, f32): VGPR0 = row K0 (lanes0-15) / K2 (lanes16-31),
    //                       VGPR1 = row K1 / K3.
    v2f wf0[16], wf1[16], wf2[16];
    #pragma unroll
    for (int s = 0; s < 16; ++s) {
        const int kr = s * 4 + 2 * half;
        wf0[s].x = W[(0 * 64 + kr    ) * 64 + n0 + l16];
        wf0[s].y = W[(0 * 64 + kr + 1) * 64 + n0 + l16];
        wf1[s].x = W[(1 * 64 + kr    ) * 64 + n0 + l16];
        wf1[s].y = W[(1 * 64 + kr + 1) * 64 + n0 + l16];
        wf2[s].x = W[(2 * 64 + kr    ) * 64 + n0 + l16];
        wf2[s].y = W[(2 * 64 + kr + 1) * 64 + n0 + l16];
    }
    const float bn = bias[n0 + l16];

    const int wavesPerGrid = gridDim.y * (blockDim.x >> 5);
    for (int tile = blockIdx.y * (blockDim.x >> 5) + wave; tile < numRowTiles;
         tile += wavesPerGrid) {
        const int row0 = tile * 16;
        // A layout (16x4, f32): lane holds row M=l16; VGPR0 = K(2*half), VGPR1 = K(2*half+1)
        const float* a0b = Tx0 + (size_t)(row0 + l16) * FEAT;
        const float* a1b = Tx1 + (size_t)(row0 + l16) * FEAT;
        const float* a2b = P2  + (size_t)(row0 + l16) * FEAT;
        v8f c = {};
        #pragma unroll
        for (int s = 0; s < 16; ++s) {
            const int kc = s * 4 + 2 * half;
            v2f a0, a1, a2;
            a0.x = a0b[kc];  a0.y = a0b[kc + 1];
            a1.x = a1b[kc];  a1.y = a1b[kc + 1];
            a2.x = 2.f * a2b[kc]     - a0.x;      // Tx2 = 2*P2 - Tx0 (fused)
            a2.y = 2.f * a2b[kc + 1] - a0.y;
            c = __builtin_amdgcn_wmma_f32_16x16x4_f32(false, a0, false, wf0[s],
                                                      (short)0, c, false, false);
            c = __builtin_amdgcn_wmma_f32_16x16x4_f32(false, a1, false, wf1[s],
                                                      (short)0, c, false, false);
            c = __builtin_amdgcn_wmma_f32_16x16x4_f32(false, a2, false, wf2[s],
                                                      (short)0, c, false, false);
        }
        // C/D layout: VGPR v -> row (v + 8*half), col l16.
        #pragma unroll
        for (int v = 0; v < 8; ++v) {
            const int r = row0 + v + 8 * half;
            float val = c[v] + bn;
            val = (val > 0.f) ? val : (expf(val) - 1.f);    // ELU
            Hout[(size_t)r * FEAT + n0 + l16] = val;
        }
    }
}

// ---------------- mean pool + linear + log_softmax (one block per graph) ----
__global__ __launch_bounds__(256)
void pool_head_kernel(const float* __restrict__ H3, const float* __restrict__ Wlin,
                      const float* __restrict__ blin, float* __restrict__ out, int N) {
    __shared__ float partial[256];
    __shared__ float g[FEAT];
    __shared__ float logits[10];
    const int b = blockIdx.x;
    const int t = threadIdx.x;
    const int h = t & 63, part = t >> 6;
    float s = 0.f;
    for (int n = part; n < N; n += 4)
        s += H3[((size_t)b * N + n) * FEAT + h];
    partial[t] = s;
    __syncthreads();
    if (part == 0)
        g[h] = (partial[h] + partial[64 + h] + partial[128 + h] + partial[192 + h]) / (float)N;
    __syncthreads();
    if (t < 10) {
        float acc = blin[t];
        for (int hh = 0; hh < FEAT; ++hh) acc += g[hh] * Wlin[hh * 10 + t];
        logits[t] = acc;
    }
    __syncthreads();
    if (t == 0) {
        float m = logits[0];
        for (int c = 1; c < 10; ++c) m = fmaxf(m, logits[c]);
        float sum = 0.f;
        for (int c = 0; c < 10; ++c) sum += expf(logits[c] - m);
        float lse = m + logf(sum);
        for (int c = 0; c < 10; ++c) out[b * 10 + c] = logits[c] - lse;
    }
}

// ---------------------------------------------------------------------------
extern "C" void kernel_launch(void* const* d_in, const int* in_sizes, int n_in,
                              void* d_out, int out_size, void* d_ws, size_t ws_size,
                              hipStream_t stream) {
    const float* x    = (const float*)d_in[0];
    const int*   ei   = (const int*)  d_in[1];
    const float* W1   = (const float*)d_in[2];
    const float* b1   = (const float*)d_in[3];
    const float* W2   = (const float*)d_in[4];
    const float* b2   = (const float*)d_in[5];
    const float* W3   = (const float*)d_in[6];
    const float* b3   = (const float*)d_in[7];
    const float* Wlin = (const float*)d_in[8];
    const float* blin = (const float*)d_in[9];
    float* out = (float*)d_out;

    const int N = in_sizes[0] / (BATCH * FEAT);   // 20000
    const int E = in_sizes[1] / 2;                // 640000
    const int* src = ei;
    const int* dst = ei + E;

    // workspace layout (floats)
    float* ws = (float*)d_ws;
    const size_t BN64 = (size_t)BATCH * N * FEAT; // 5.12M floats
    float* bufA = ws;
    float* bufB = ws + BN64;
    float* bufC = ws + 2 * BN64;
    float* bufD = ws + 3 * BN64;
    float* deg  = ws + 4 * BN64;
    float* wbuf = deg + (((size_t)N + 63) / 64) * 64;

    const int TPB = 256;
    const int n4     = (int)(BN64 / 4);
    const int zgrid  = (n4 + TPB - 1) / TPB;
    const long long propTotal = (long long)E * BATCH * 16;
    const int pgrid  = (int)((propTotal + TPB - 1) / TPB);
    const int numRowTiles = (BATCH * N) / 16;     // 5000
    dim3 gGrid(4, 250);                           // 4 col-tiles x row chunks
    dim3 gBlock(128);                             // 4 waves/block, 5 row tiles/wave

    // --- edge weights ---
    fill0_scalar<<<(N + TPB - 1) / TPB, TPB, 0, stream>>>(deg, N);
    deg_kernel  <<<(E + TPB - 1) / TPB, TPB, 0, stream>>>(src, dst, deg, E);
    dis_kernel  <<<(N + TPB - 1) / TPB, TPB, 0, stream>>>(deg, N);
    edgew_kernel<<<(E + TPB - 1) / TPB, TPB, 0, stream>>>(src, dst, deg, wbuf, E);

    // --- h0 = x^T -> bufA ---
    transpose_kernel<<<dim3((N + 31) / 32, BATCH), 256, 0, stream>>>(x, bufA, N);

    // --- layer 1: Tx0=A, Tx1=B, P2=C, out=D ---
    fill0_vec4<<<zgrid, TPB, 0, stream>>>((float4*)bufB, n4);
    propagate_kernel<<<pgrid, TPB, 0, stream>>>(bufA, src, dst, wbuf, bufB, N, E);
    fill0_vec4<<<zgrid, TPB, 0, stream>>>((float4*)bufC, n4);
    propagate_kernel<<<pgrid, TPB, 0, stream>>>(bufB, src, dst, wbuf, bufC, N, E);
    cheb_layer_kernel<<<gGrid, gBlock, 0, stream>>>(bufA, bufB, bufC, W1, b1, bufD, numRowTiles);

    // --- layer 2: Tx0=D, out=A ---
    fill0_vec4<<<zgrid, TPB, 0, stream>>>((float4*)bufB, n4);
    propagate_kernel<<<pgrid, TPB, 0, stream>>>(bufD, src, dst, wbuf, bufB, N, E);
    fill0_vec4<<<zgrid, TPB, 0, stream>>>((float4*)bufC, n4);
    propagate_kernel<<<pgrid, TPB, 0, stream>>>(bufB, src, dst, wbuf, bufC, N, E);
    cheb_layer_kernel<<<gGrid, gBlock, 0, stream>>>(bufD, bufB, bufC, W2, b2, bufA, numRowTiles);

    // --- layer 3: Tx0=A, out=D ---
    fill0_vec4<<<zgrid, TPB, 0, stream>>>((float4*)bufB, n4);
    propagate_kernel<<<pgrid, TPB, 0, stream>>>(bufA, src, dst, wbuf, bufB, N, E);
    fill0_vec4<<<zgrid, TPB, 0, stream>>>((float4*)bufC, n4);
    propagate_kernel<<<pgrid, TPB, 0, stream>>>(bufB, src, dst, wbuf, bufC, N, E);
    cheb_layer_kernel<<<gGrid, gBlock, 0, stream>>>(bufA, bufB, bufC, W3, b3, bufD, numRowTiles);

    // --- pool + linear + log_softmax ---
    pool_head_kernel<<<BATCH, 256, 0, stream>>>(bufD, Wlin, blin, out, N);
}